// OBM_GENConv_65652870087447
// MI455X (gfx1250) — compile-verified
//
#include <hip/hip_runtime.h>
#include <math.h>

// ---------------------------------------------------------------------------
// GENConv (3 layers) for MI455X / gfx1250, wave32.
//  - GEMMs:  V_WMMA_F32_16X16X4_F32, one wave per 16x16 tile; kernels are
//            templated on (N,K) so the k-loop unrolls into batched-load /
//            back-to-back WMMA streams. Weight K-slabs are staged into LDS by
//            the Tensor Data Mover (tensor_load_to_lds + s_wait_tensorcnt);
//            B-fragments come from ds_load_b64.
//  - Segment softmax: 3 streaming passes over e (non-temporal), gathers and
//            atomics land in the 192MB L2 (dst accumulators are 25.6MB).
//  - msg = relu(..)+1e-7 > 0 always => segment max via integer atomicMax on
//            float bits, init 0 (exactly reproduces the empty-segment fixup).
//
// Workspace (floats): h | hsrc | hdst | aggr | m | s | e | stats
//   (m,s contiguous and reused as the N*256 MLP hidden buffer y)
// ---------------------------------------------------------------------------

typedef __attribute__((ext_vector_type(2))) float v2f;
typedef __attribute__((ext_vector_type(4))) float v4f;
typedef __attribute__((ext_vector_type(8))) float v8f;
typedef __attribute__((ext_vector_type(4))) unsigned int u32x4;
typedef __attribute__((ext_vector_type(8))) int i32x8;
typedef __attribute__((ext_vector_type(4))) int i32x4;

#define NN 50000
#define NE 800000
#define HID 128

#if defined(__has_builtin)
#if __has_builtin(__builtin_amdgcn_tensor_load_to_lds) && \
    __has_builtin(__builtin_amdgcn_s_wait_tensorcnt)
#define USE_TDM 1
#endif
#endif

// ---------------------------------------------------------------------------
// C[M,N] = act(A[M,K] @ W[N,K]^T + bias[N]); one wave per 16x16 tile.
// W is staged into LDS in [N, KS] slabs by the TDM (one DMA per block/slab).
// A-frag: lane<16 holds row (lane&15), k=k0+0/1; lane>=16 holds k=k0+2/3.
// D: VGPR j -> row tm*16 + j + 8*(lane>>4), col tn*16 + (lane&15).
// M%16==0; N, K compile-time; KS divides K for all instantiations used.
// ---------------------------------------------------------------------------
template <int N, int K>
__global__ void wmma_gemm_nt(const float* __restrict__ A,
                             const float* __restrict__ W,
                             const float* __restrict__ bias,
                             float* __restrict__ C,
                             int M, int relu)
{
    constexpr int KS = ((16384 / N) < K) ? (16384 / N) : K;   // slab width
    __shared__ float ldsW[N * KS];                            // <= 64 KB

    const int lane   = threadIdx.x & 31;
    const int wid    = (int)((blockIdx.x * blockDim.x + threadIdx.x) >> 5);
    const int tilesN = N >> 4;
    const int tilesM = M >> 4;
    int tm = wid / tilesN;
    int tn = wid % tilesN;
    const bool valid = (tm < tilesM);        // wave-uniform
    if (!valid) { tm = 0; tn = 0; }          // clamp: keep barriers matched

    const int  half = lane >> 4;
    const int  l15  = lane & 15;
    const int  kb   = half << 1;             // 0 or 2
    const size_t arow = (size_t)(tm * 16 + l15) * (size_t)K;
    const int    brow = tn * 16 + l15;

    v8f acc = {};
    for (int ks = 0; ks < K; ks += KS) {
        // ---- stage W[:, ks:ks+KS] -> ldsW (packed rows of KS floats) ----
#ifdef USE_TDM
        if (threadIdx.x < 32) {              // one TDM issue per block
            unsigned long long ga =
                (unsigned long long)(uintptr_t)W + (unsigned long long)ks * 4ull;
            unsigned ldsAddr = (unsigned)(uintptr_t)&ldsW[0];
            constexpr unsigned td0 = (unsigned)KS;   // tensor/tile dim0
            constexpr unsigned td1 = (unsigned)N;    // tile dim1 = rows
            constexpr unsigned st0 = (unsigned)K;    // row stride (elements)

            u32x4 g0;
            g0.x = 1u;                                            // count=1
            g0.y = ldsAddr;                                       // lds_addr
            g0.z = (unsigned)(ga & 0xffffffffull);                // gaddr lo
            g0.w = (unsigned)((ga >> 32) & 0x1ffffffull) | (2u << 30); // type=2

            i32x8 g1;
            g1[0] = (int)(2u << 16);                              // data_size=4B
            g1[1] = (int)((td0 & 0xffffu) << 16);                 // tensor_dim0[15:0]
            g1[2] = (int)(((td0 >> 16) & 0xffffu) |               // tensor_dim0[31:16]
                          ((td1 & 0xffffu) << 16));               // tensor_dim1[15:0]
            g1[3] = (int)(((td1 >> 16) & 0xffffu) |               // tensor_dim1[31:16]
                          ((td0 & 0xffffu) << 16));               // tile_dim0 = KS
            g1[4] = (int)(td1 & 0xffffu);                         // tile_dim1=N, tile_dim2=0
            g1[5] = (int)st0;                                     // dim0_stride lo
            g1[6] = 0;                                            // dim0_stride hi
            g1[7] = 0;

            i32x4 gz4 = {0, 0, 0, 0};
            i32x8 gz8 = {0, 0, 0, 0, 0, 0, 0, 0};
            __builtin_amdgcn_tensor_load_to_lds(g0, g1, gz4, gz4, gz8, 0);
            __builtin_amdgcn_s_wait_tensorcnt(0);
        }
#else
        for (int i = threadIdx.x; i < N * KS; i += (int)blockDim.x) {
            int r = i / KS, c = i - r * KS;
            ldsW[i] = W[(size_t)r * K + ks + c];
        }
#endif
        __syncthreads();

        // ---- WMMA over the slab: A from global, B from LDS ----
        const float* bp = ldsW + (size_t)brow * KS;
#pragma unroll 8
        for (int k0 = 0; k0 < KS; k0 += 4) {
            v2f a = *(const v2f*)(A + arow + ks + k0 + kb);
            v2f b = *(const v2f*)(bp + k0 + kb);
            acc = __builtin_amdgcn_wmma_f32_16x16x4_f32(
                /*neg_a=*/false, a, /*neg_b=*/false, b,
                /*c_mod=*/(short)0, acc, /*reuse_a=*/false, /*reuse_b=*/false);
        }
        __syncthreads();                     // before next slab overwrites LDS
    }

    if (valid) {
        const int   col   = tn * 16 + l15;
        const float bc    = bias[col];
        const int   rbase = tm * 16 + (half << 3);
#pragma unroll
        for (int j = 0; j < 8; ++j) {
            float v = acc[j] + bc;
            if (relu) v = fmaxf(v, 0.0f);
            C[(size_t)(rbase + j) * (size_t)N + col] = v;
        }
    }
}

// aggr = h_dst ; m = 0 (msg > 0 always => matches empty-segment fixup) ; s = 0
__global__ void init_seg(const float* __restrict__ hdst,
                         float* __restrict__ aggr,
                         float* __restrict__ m,
                         float* __restrict__ s)
{
    size_t i = (size_t)blockIdx.x * blockDim.x + threadIdx.x;
    if (i < (size_t)NN * HID) {
        aggr[i] = hdst[i];
        m[i]    = 0.0f;
        s[i]    = 0.0f;
    }
}

// One wave per edge, each lane owns 4 channels (float4).
// pass 1: m[dst]    = max(m[dst], msg)
// pass 2: s[dst]   += exp(msg - m[dst])
// pass 3: aggr[dst]+= msg * exp(msg-m[dst]) / (s[dst] + 1e-16)
__global__ void edge_pass(const int* __restrict__ srcI,
                          const int* __restrict__ dstI,
                          const float* __restrict__ e,
                          const float* __restrict__ hsrc,
                          const float* __restrict__ m,
                          const float* __restrict__ s,
                          float* __restrict__ outp,
                          int pass)
{
    const int edge = (int)((blockIdx.x * blockDim.x + threadIdx.x) >> 5);
    if (edge >= NE) return;
    const int lane = threadIdx.x & 31;
    const int c    = lane << 2;
    const int u    = srcI[edge];
    const int v    = dstI[edge];

    v4f ev = __builtin_nontemporal_load((const v4f*)(e + (size_t)edge * HID + c));
    v4f hv = *(const v4f*)(hsrc + (size_t)u * HID + c);
    v4f msg;
    msg.x = fmaxf(ev.x + hv.x, 0.0f) + 1e-7f;
    msg.y = fmaxf(ev.y + hv.y, 0.0f) + 1e-7f;
    msg.z = fmaxf(ev.z + hv.z, 0.0f) + 1e-7f;
    msg.w = fmaxf(ev.w + hv.w, 0.0f) + 1e-7f;

    float* orow = outp + (size_t)v * HID + c;

    if (pass == 1) {
        atomicMax((int*)(orow + 0), __float_as_int(msg.x));
        atomicMax((int*)(orow + 1), __float_as_int(msg.y));
        atomicMax((int*)(orow + 2), __float_as_int(msg.z));
        atomicMax((int*)(orow + 3), __float_as_int(msg.w));
    } else {
        v4f mv = *(const v4f*)(m + (size_t)v * HID + c);
        v4f p;
        p.x = __expf(msg.x - mv.x);
        p.y = __expf(msg.y - mv.y);
        p.z = __expf(msg.z - mv.z);
        p.w = __expf(msg.w - mv.w);
        if (pass == 2) {
            __hip_atomic_fetch_add(orow + 0, p.x, __ATOMIC_RELAXED, __HIP_MEMORY_SCOPE_AGENT);
            __hip_atomic_fetch_add(orow + 1, p.y, __ATOMIC_RELAXED, __HIP_MEMORY_SCOPE_AGENT);
            __hip_atomic_fetch_add(orow + 2, p.z, __ATOMIC_RELAXED, __HIP_MEMORY_SCOPE_AGENT);
            __hip_atomic_fetch_add(orow + 3, p.w, __ATOMIC_RELAXED, __HIP_MEMORY_SCOPE_AGENT);
        } else {
            v4f sv = *(const v4f*)(s + (size_t)v * HID + c);
            __hip_atomic_fetch_add(orow + 0, msg.x * p.x / (sv.x + 1e-16f), __ATOMIC_RELAXED, __HIP_MEMORY_SCOPE_AGENT);
            __hip_atomic_fetch_add(orow + 1, msg.y * p.y / (sv.y + 1e-16f), __ATOMIC_RELAXED, __HIP_MEMORY_SCOPE_AGENT);
            __hip_atomic_fetch_add(orow + 2, msg.z * p.z / (sv.z + 1e-16f), __ATOMIC_RELAXED, __HIP_MEMORY_SCOPE_AGENT);
            __hip_atomic_fetch_add(orow + 3, msg.w * p.w / (sv.w + 1e-16f), __ATOMIC_RELAXED, __HIP_MEMORY_SCOPE_AGENT);
        }
    }
}

__global__ void bn_zero(float* __restrict__ stats)
{
    int t = threadIdx.x;
    stats[t] = 0.0f;
    stats[256 + t] = 0.0f;
}

// thread t owns column t (coalesced across 256-wide rows); block folds a row
// chunk's partial sum / sumsq into stats with f32 atomics.
__global__ void bn_partial(const float* __restrict__ y, float* __restrict__ stats,
                           int rowsPerBlock)
{
    int c  = threadIdx.x;
    int r0 = blockIdx.x * rowsPerBlock;
    int r1 = r0 + rowsPerBlock; if (r1 > NN) r1 = NN;
    float sm = 0.0f, sq = 0.0f;
    for (int r = r0; r < r1; ++r) {
        float v = y[(size_t)r * 256 + c];
        sm += v; sq += v * v;
    }
    __hip_atomic_fetch_add(&stats[c],       sm, __ATOMIC_RELAXED, __HIP_MEMORY_SCOPE_AGENT);
    __hip_atomic_fetch_add(&stats[256 + c], sq, __ATOMIC_RELAXED, __HIP_MEMORY_SCOPE_AGENT);
}

__global__ void bn_finalize(const float* __restrict__ stats,
                            const float* __restrict__ g, const float* __restrict__ b,
                            float* __restrict__ ss)
{
    int c = threadIdx.x;
    float mean = stats[c] * (1.0f / NN);
    float var  = stats[256 + c] * (1.0f / NN) - mean * mean;   // ddof=0 (jnp.var)
    float inv  = rsqrtf(var + 1e-5f);
    float sc   = g[c] * inv;
    ss[c]       = sc;
    ss[256 + c] = b[c] - mean * sc;
}

__global__ void bn_apply_relu(float* __restrict__ y, const float* __restrict__ ss)
{
    size_t i = (size_t)blockIdx.x * blockDim.x + threadIdx.x;
    if (i < (size_t)NN * 256) {
        int c = (int)(i & 255);
        y[i] = fmaxf(y[i] * ss[c] + ss[256 + c], 0.0f);
    }
}

// out[node] = dot(h[node,:128], w) + b ; one wave per node, shuffle reduce.
__global__ void head_kernel(const float* __restrict__ h, const float* __restrict__ w,
                            const float* __restrict__ b, float* __restrict__ out)
{
    const int node = (int)((blockIdx.x * blockDim.x + threadIdx.x) >> 5);
    if (node >= NN) return;
    const int lane = threadIdx.x & 31;
    v4f hv = *(const v4f*)(h + (size_t)node * HID + lane * 4);
    v4f wv = *(const v4f*)(w + lane * 4);
    float sum = hv.x * wv.x + hv.y * wv.y + hv.z * wv.z + hv.w * wv.w;
#pragma unroll
    for (int off = 16; off > 0; off >>= 1) sum += __shfl_down(sum, off, 32);
    if (lane == 0) out[node] = sum + b[0];
}

// ---------------------------------------------------------------------------
template <int N, int K>
static inline void launch_gemm(const float* A, const float* W, const float* bias,
                               float* C, int M, int relu, hipStream_t st)
{
    int tiles  = (M >> 4) * (N >> 4);
    int blocks = (tiles + 7) / 8;                    // 8 waves / 256-thread block
    hipLaunchKernelGGL((wmma_gemm_nt<N, K>), dim3(blocks), dim3(256), 0, st,
                       A, W, bias, C, M, relu);
}

extern "C" void kernel_launch(void* const* d_in, const int* in_sizes, int n_in,
                              void* d_out, int out_size, void* d_ws, size_t ws_size,
                              hipStream_t stream)
{
    (void)n_in; (void)out_size; (void)ws_size;
    auto f = [&](int i) { return (const float*)d_in[i]; };

    const float *xP, *eaP, *hw, *hb;
    const int   *eiP;
    const float *lsW, *lsB, *ldW, *ldB;
    const float *leW[3], *leB[3], *w1[3], *b1[3], *g[3], *bb[3], *w2[3], *b2[3];

    if (in_sizes[0] == NN * 32) {
        // setup_inputs() insertion order
        xP = f(0); eiP = (const int*)d_in[1]; eaP = f(2);
        lsW = f(3);  lsB = f(4);  ldW = f(5);  ldB = f(6);
        leW[0] = f(7);  leB[0] = f(8);  w1[0] = f(9);  b1[0] = f(10);
        g[0] = f(11); bb[0] = f(12); w2[0] = f(13); b2[0] = f(14);
        int base = 15;
        for (int l = 1; l < 3; ++l, base += 8) {
            leW[l] = f(base);     leB[l] = f(base + 1);
            w1[l]  = f(base + 2); b1[l]  = f(base + 3);
            g[l]   = f(base + 4); bb[l]  = f(base + 5);
            w2[l]  = f(base + 6); b2[l]  = f(base + 7);
        }
        hw = f(31); hb = f(32);
    } else {
        // jax pytree (sorted dict keys) order
        eaP = f(0); eiP = (const int*)d_in[1]; hb = f(2); hw = f(3);
        bb[0] = f(4);  g[0]  = f(5);  ldB = f(6);  ldW = f(7);
        leB[0] = f(8); leW[0] = f(9); lsB = f(10); lsW = f(11);
        b1[0] = f(12); b2[0] = f(13); w1[0] = f(14); w2[0] = f(15);
        int base = 16;
        for (int l = 1; l < 3; ++l, base += 8) {
            bb[l]  = f(base);     g[l]   = f(base + 1);
            leB[l] = f(base + 2); leW[l] = f(base + 3);
            b1[l]  = f(base + 4); b2[l]  = f(base + 5);
            w1[l]  = f(base + 6); w2[l]  = f(base + 7);
        }
        xP = f(32);
    }

    // workspace carve-up
    float* ws    = (float*)d_ws;
    float* h     = ws;
    float* hsrc  = h    + (size_t)NN * HID;
    float* hdst  = hsrc + (size_t)NN * HID;
    float* aggr  = hdst + (size_t)NN * HID;
    float* m     = aggr + (size_t)NN * HID;
    float* s     = m    + (size_t)NN * HID;   // y = m..s as contiguous N*256
    float* e     = s    + (size_t)NN * HID;
    float* stats = e    + (size_t)NE * HID;   // 512 stats + 512 scale/shift
    float* y     = m;
    float* ssbuf = stats + 512;

    const int* srcI = eiP;
    const int* dstI = eiP + NE;

    const int initBlocks = ((NN * HID) + 255) / 256;
    const int edgeBlocks = (NE * 32) / 256;           // one wave per edge
    const int bnRows     = 256;
    const int bnBlocks   = (NN + bnRows - 1) / bnRows;
    const int applyBlk   = ((NN * 256) + 255) / 256;

    for (int l = 0; l < 3; ++l) {
        const float *HS, *HD;
        if (l == 0) {
            launch_gemm<128, 32>(xP, lsW, lsB, hsrc, NN, 0, stream);
            launch_gemm<128, 32>(xP, ldW, ldB, hdst, NN, 0, stream);
            HS = hsrc; HD = hdst;
        } else {
            HS = h; HD = h;
        }

        // edge linear: e = edge_attr @ W_edge^T + b
        launch_gemm<128, 16>(eaP, leW[l], leB[l], e, NE, 0, stream);

        // segment softmax aggregation
        hipLaunchKernelGGL(init_seg, dim3(initBlocks), dim3(256), 0, stream,
                           HD, aggr, m, s);
        hipLaunchKernelGGL(edge_pass, dim3(edgeBlocks), dim3(256), 0, stream,
                           srcI, dstI, e, HS, m, s, m, 1);
        hipLaunchKernelGGL(edge_pass, dim3(edgeBlocks), dim3(256), 0, stream,
                           srcI, dstI, e, HS, m, s, s, 2);
        hipLaunchKernelGGL(edge_pass, dim3(edgeBlocks), dim3(256), 0, stream,
                           srcI, dstI, e, HS, m, s, aggr, 3);

        // MLP: Linear(128->256) -> BN -> ReLU -> Linear(256->128) (+ layer ReLU)
        launch_gemm<256, 128>(aggr, w1[l], b1[l], y, NN, 0, stream);
        hipLaunchKernelGGL(bn_zero, dim3(1), dim3(256), 0, stream, stats);
        hipLaunchKernelGGL(bn_partial, dim3(bnBlocks), dim3(256), 0, stream,
                           y, stats, bnRows);
        hipLaunchKernelGGL(bn_finalize, dim3(1), dim3(256), 0, stream,
                           stats, g[l], bb[l], ssbuf);
        hipLaunchKernelGGL(bn_apply_relu, dim3(applyBlk), dim3(256), 0, stream,
                           y, ssbuf);
        launch_gemm<128, 256>(y, w2[l], b2[l], h, NN, 1, stream);
    }

    hipLaunchKernelGGL(head_kernel, dim3((NN * 32 + 255) / 256), dim3(256), 0, stream,
                       h, hw, hb, (float*)d_out);
}